// ActorNetwork_64321430225479
// MI455X (gfx1250) — compile-verified
//
#include <hip/hip_runtime.h>
#include <hip/hip_bf16.h>

// ----------------------------------------------------------------------------
// ActorNetwork (2-layer GCN -> mean pool -> LSTM step -> FC -> log_softmax)
// MI455X / gfx1250 implementation.
//
// Roofline: edge aggregation dominates (L2-resident, ~12.8MB feature maps in
// 192MB L2); node GEMMs use fp32 WMMA (V_WMMA_F32_16X16X4_F32) since fp32 is
// required and the GEMM FLOPs are tiny relative to the gather/scatter traffic.
// ----------------------------------------------------------------------------

typedef __attribute__((ext_vector_type(2))) float v2f;
typedef __attribute__((ext_vector_type(8))) float v8f;

#define CHID 32

// ---------------- init: B=0 (agg target), deg=1 (self loop), pooled=0 -------
__global__ void k_init(float* B, float* deg, float* pooled, int n_nodes) {
  int idx = blockIdx.x * blockDim.x + threadIdx.x;
  int total = n_nodes * CHID;
  if (idx < total) B[idx] = 0.0f;
  if (idx < n_nodes) deg[idx] = 1.0f;   // self-loop contribution
  if (idx < CHID) pooled[idx] = 0.0f;
}

// ---------------- deg[c] += 1 per edge --------------------------------------
__global__ void k_degree(const int* __restrict__ col, float* deg, int n_edges) {
  int e = blockIdx.x * blockDim.x + threadIdx.x;
  if (e < n_edges) atomicAdd(&deg[col[e]], 1.0f);
}

__global__ void k_dinv(const float* __restrict__ deg, float* __restrict__ dinv,
                       int n_nodes) {
  int i = blockIdx.x * blockDim.x + threadIdx.x;
  if (i < n_nodes) {
    float d = deg[i];
    dinv[i] = (d > 0.0f) ? rsqrtf(d) : 0.0f;
  }
}

// ---------------- fp32 WMMA GEMM: H[n,32] = X[n,K] @ W[K,32] ----------------
// One wave per 16-row tile. Two 16x16 f32 accumulators cover N=32.
// A layout (16x4 f32): lanes 0-15 hold M=lane, K={k,k+1}; lanes 16-31 K={k+2,k+3}.
// B layout (4x16 f32): V0 = row K (lanes0-15) / K+2 (lanes16-31), V1 = K+1/K+3.
// C/D layout: VGPR j -> M=j (lanes0-15) / M=j+8 (lanes16-31), N=lane&15.
__global__ __launch_bounds__(128)
void k_gemm_wmma(const float* X, const float* W, float* H, int n_nodes, int K) {
  const int lane = threadIdx.x & 31;
  const int wave = threadIdx.x >> 5;
  const int tiles = (n_nodes + 15) >> 4;
  const int tile = blockIdx.x * 4 + wave;
  if (tile >= tiles) return;                 // wave-uniform exit

  const int half = lane >> 4;                // 0 or 1
  const int l15  = lane & 15;
  int row = tile * 16 + l15;
  if (row >= n_nodes) row = n_nodes - 1;     // clamp loads, keep EXEC all-1s
  const int kh = half * 2;

  v8f c0 = {};
  v8f c1 = {};
  for (int k = 0; k < K; k += 4) {
    v2f a;
    a.x = X[row * K + k + kh];
    a.y = X[row * K + k + kh + 1];
    v2f b0, b1;
    b0.x = W[(k + kh)     * CHID + l15];
    b0.y = W[(k + kh + 1) * CHID + l15];
    b1.x = W[(k + kh)     * CHID + 16 + l15];
    b1.y = W[(k + kh + 1) * CHID + 16 + l15];
    c0 = __builtin_amdgcn_wmma_f32_16x16x4_f32(false, a, false, b0,
                                               (short)0, c0, false, false);
    c1 = __builtin_amdgcn_wmma_f32_16x16x4_f32(false, a, false, b1,
                                               (short)0, c1, false, false);
  }
  const int mbase = tile * 16 + half * 8;
#pragma unroll
  for (int j = 0; j < 8; ++j) {
    int m = mbase + j;
    if (m < n_nodes) {
      H[m * CHID + l15]      = c0[j];
      H[m * CHID + 16 + l15] = c1[j];
    }
  }
}

// ---------------- edge gather+scale+scatter-add (L2-resident) ---------------
__global__ void k_edge_agg(const int* __restrict__ row, const int* __restrict__ col,
                           const float* __restrict__ dinv, const float* __restrict__ H,
                           float* AGG, int n_edges) {
  int e = blockIdx.x * blockDim.x + threadIdx.x;
  if (e >= n_edges) return;
  int r = row[e], c = col[e];
  float norm = dinv[r] * dinv[c];
  const float4* hs = (const float4*)(H + (size_t)r * CHID);
  float* out = AGG + (size_t)c * CHID;
#pragma unroll
  for (int i = 0; i < 8; ++i) {
    float4 v = hs[i];
    atomicAdd(out + 4 * i + 0, v.x * norm);
    atomicAdd(out + 4 * i + 1, v.y * norm);
    atomicAdd(out + 4 * i + 2, v.z * norm);
    atomicAdd(out + 4 * i + 3, v.w * norm);
  }
}

// ---------------- layer-1 post: X1 = relu(agg + h*dinv^2 + b); clear agg ----
__global__ void k_post(const float* AGG, float* H, float* AGGclr,
                       const float* __restrict__ dinv, const float* __restrict__ b,
                       int n_nodes) {
  int idx = blockIdx.x * blockDim.x + threadIdx.x;
  if (idx >= n_nodes * CHID) return;
  int node = idx >> 5, ch = idx & 31;
  float di = dinv[node];
  float val = AGG[idx] + H[idx] * di * di + b[ch];
  H[idx] = fmaxf(val, 0.0f);
  AGGclr[idx] = 0.0f;  // same buffer as AGG; thread-local read-then-clear
}

// ---------------- layer-2 post + mean-pool partials -------------------------
__global__ __launch_bounds__(256)
void k_post_pool(const float* AGG, const float* H, const float* __restrict__ dinv,
                 const float* __restrict__ b, float* pooled, int n_nodes) {
  __shared__ float s[CHID];
  int t = threadIdx.x;
  if (t < CHID) s[t] = 0.0f;
  __syncthreads();
  int idx = blockIdx.x * blockDim.x + t;
  if (idx < n_nodes * CHID) {
    int node = idx >> 5, ch = idx & 31;
    float di = dinv[node];
    float val = fmaxf(AGG[idx] + H[idx] * di * di + b[ch], 0.0f);
    atomicAdd(&s[ch], val);                    // ds_add_f32
  }
  __syncthreads();
  if (t < CHID) atomicAdd(&pooled[t], s[t]);   // global_atomic_add_f32
}

// ---------------- tail: mean, concat, LSTM step, FC, log_softmax ------------
__global__ __launch_bounds__(256)
void k_tail(const float* pooled, const float* x_state, const float* h0,
            const float* c0, const float* W_ih, const float* W_hh,
            const float* b_ih, const float* b_hh, const float* W_fc,
            const float* b_fc, float* out, int n_nodes) {
  __shared__ float xcomb[96];
  __shared__ float gates[1024];
  __shared__ float hnew[256];
  __shared__ float logits[64];
  __shared__ float red[2];
  int t = threadIdx.x;
  if (t < CHID) xcomb[t] = pooled[t] / (float)n_nodes;
  else if (t < 96) xcomb[t] = x_state[t - CHID];
  __syncthreads();
  for (int k = t; k < 1024; k += 256) {
    float g = b_ih[k] + b_hh[k];
    const float* wi = W_ih + k * 96;
    for (int j = 0; j < 96; ++j) g += xcomb[j] * wi[j];
    const float* wh = W_hh + k * 256;
    for (int j = 0; j < 256; ++j) g += h0[j] * wh[j];
    gates[k] = g;
  }
  __syncthreads();
  {
    float ig = 1.0f / (1.0f + expf(-gates[t]));
    float fg = 1.0f / (1.0f + expf(-gates[256 + t]));
    float gg = tanhf(gates[512 + t]);
    float og = 1.0f / (1.0f + expf(-gates[768 + t]));
    float cn = fg * c0[t] + ig * gg;
    float hn = og * tanhf(cn);
    out[64 + t] = hn;    // h_new
    out[320 + t] = cn;   // c_new
    hnew[t] = hn;
  }
  __syncthreads();
  if (t < 64) {
    float l = b_fc[t];
    for (int j = 0; j < 256; ++j) l += hnew[j] * W_fc[j * 64 + t];
    logits[t] = l;
  }
  __syncthreads();
  if (t == 0) {
    float m = logits[0];
    for (int j = 1; j < 64; ++j) m = fmaxf(m, logits[j]);
    float s = 0.0f;
    for (int j = 0; j < 64; ++j) s += expf(logits[j] - m);
    red[0] = m;
    red[1] = logf(s);
  }
  __syncthreads();
  if (t < 64) out[t] = logits[t] - red[0] - red[1];
}

// ----------------------------------------------------------------------------
extern "C" void kernel_launch(void* const* d_in, const int* in_sizes, int n_in,
                              void* d_out, int out_size, void* d_ws, size_t ws_size,
                              hipStream_t stream) {
  const float* X       = (const float*)d_in[0];
  const int*   ei      = (const int*)d_in[1];
  const float* x_state = (const float*)d_in[2];
  const float* h0      = (const float*)d_in[3];
  const float* c0      = (const float*)d_in[4];
  const float* W1      = (const float*)d_in[5];
  const float* b1      = (const float*)d_in[6];
  const float* W2      = (const float*)d_in[7];
  const float* b2      = (const float*)d_in[8];
  const float* W_ih    = (const float*)d_in[9];
  const float* W_hh    = (const float*)d_in[10];
  const float* b_ih    = (const float*)d_in[11];
  const float* b_hh    = (const float*)d_in[12];
  const float* W_fc    = (const float*)d_in[13];
  const float* b_fc    = (const float*)d_in[14];
  float* out = (float*)d_out;

  const int n_nodes = in_sizes[0] / 64;
  const int n_edges = in_sizes[1] / 2;
  const int* row = ei;
  const int* col = ei + n_edges;

  float* ws = (float*)d_ws;
  float* A      = ws;                          // n*32: h1 -> x1 -> h2 (in-place)
  float* B      = A + (size_t)n_nodes * CHID;  // n*32: agg target (both layers)
  float* deg    = B + (size_t)n_nodes * CHID;  // n
  float* dinv   = deg + n_nodes;               // n
  float* pooled = dinv + n_nodes;              // 32

  const int blk = 256;
  const int elems = n_nodes * CHID;
  const int eg = (elems + blk - 1) / blk;
  const int ng = (n_nodes + blk - 1) / blk;
  const int edg = (n_edges + blk - 1) / blk;
  const int tiles = (n_nodes + 15) / 16;
  const int ggrid = (tiles + 3) / 4;           // 4 waves (tiles) per block

  k_init<<<eg, blk, 0, stream>>>(B, deg, pooled, n_nodes);
  k_degree<<<edg, blk, 0, stream>>>(col, deg, n_edges);
  k_dinv<<<ng, blk, 0, stream>>>(deg, dinv, n_nodes);

  // Layer 1
  k_gemm_wmma<<<ggrid, 128, 0, stream>>>(X, W1, A, n_nodes, 64);
  k_edge_agg<<<edg, blk, 0, stream>>>(row, col, dinv, A, B, n_edges);
  k_post<<<eg, blk, 0, stream>>>(B, A, B, dinv, b1, n_nodes);

  // Layer 2 (GEMM in-place on A; B was re-zeroed by k_post)
  k_gemm_wmma<<<ggrid, 128, 0, stream>>>(A, W2, A, n_nodes, 32);
  k_edge_agg<<<edg, blk, 0, stream>>>(row, col, dinv, A, B, n_edges);
  k_post_pool<<<eg, blk, 0, stream>>>(B, A, dinv, b2, pooled, n_nodes);

  // Tail (single block)
  k_tail<<<1, 256, 0, stream>>>(pooled, x_state, h0, c0, W_ih, W_hh, b_ih, b_hh,
                                W_fc, b_fc, out, n_nodes);
}